// Attention_72791105732731
// MI455X (gfx1250) — compile-verified
//
#include <hip/hip_runtime.h>
#include <math.h>

#define S_DIM 16384
#define H_DIM 4096

typedef __attribute__((ext_vector_type(2))) float v2f;
typedef __attribute__((ext_vector_type(4))) float v4f;
typedef __attribute__((ext_vector_type(8))) float v8f;

// ---------------------------------------------------------------------------
// Column reduction via WMMA f32 16x16x4:
//   out[jbase + n] += sum_{r in chunk} w[r] * M[r*H_DIM + jbase + n]
// One wave per (16-column group, row chunk). ld == H_DIM (compile-time) so the
// inner loop is pure running-pointer + immediate-offset loads.
//  A (16x4): w[s0..s0+3] broadcast to all 16 M-rows.
//            slot (vgpr p, lane-half h) -> K = p + 2h   (ISA 32-bit A layout)
//  B (4x16): B[k][n] = M[(s0+k)*H + jbase + n], slot (vgpr q, lane-half h) -> K = q + 2h
//  D: all 16 rows identical; row 0 (acc[0], lanes 0..15) = 16 partial col sums.
// Two accumulators break the WMMA->WMMA RAW chain (2x ILP).
// ---------------------------------------------------------------------------
__global__ void colreduce_wmma_kernel(const float* __restrict__ M,
                                      const float* __restrict__ w,
                                      float* __restrict__ out,
                                      int rows_per_chunk) {
  const int lane  = threadIdx.x;       // 0..31, one wave per block
  const int hi    = lane >> 4;         // lane-half
  const int n     = lane & 15;         // column within 16-group
  const int jbase = blockIdx.x * 16;
  const int r0    = blockIdx.y * rows_per_chunk;

  // Running pointers for this lane-half (handles rows ra = s0 + 2*hi, ra+1).
  const float* bp = M + (size_t)(r0 + hi * 2) * H_DIM + jbase + n;
  const float* ap = w + r0 + hi * 2;

  v8f acc0 = {0.f, 0.f, 0.f, 0.f, 0.f, 0.f, 0.f, 0.f};
  v8f acc1 = acc0;

  const int iters = rows_per_chunk / 8;   // 8 rows per iteration
  for (int it = 0; it < iters; ++it) {
    v2f a0, b0, a1, b1;
    {
      const float2 aw = *(const float2*)ap;           // w[ra], w[ra+1]
      a0.x = aw.x; a0.y = aw.y;
    }
    b0.x = __builtin_nontemporal_load(bp);            // M[ra    ][j]
    b0.y = __builtin_nontemporal_load(bp + H_DIM);    // M[ra + 1][j]
    acc0 = __builtin_amdgcn_wmma_f32_16x16x4_f32(
        false, a0, false, b0, (short)0, acc0, false, false);

    {
      const float2 aw = *(const float2*)(ap + 4);     // w[ra+4], w[ra+5]
      a1.x = aw.x; a1.y = aw.y;
    }
    b1.x = __builtin_nontemporal_load(bp + 4 * H_DIM);
    b1.y = __builtin_nontemporal_load(bp + 5 * H_DIM);
    acc1 = __builtin_amdgcn_wmma_f32_16x16x4_f32(
        false, a1, false, b1, (short)0, acc1, false, false);

    ap += 8;
    bp += (size_t)8 * H_DIM;
  }

  if (lane < 16) atomicAdd(&out[jbase + lane], acc0[0] + acc1[0]);
}

// ---------------------------------------------------------------------------
// score[row] = dot(hs[row, :], v)  — one wave per row, float4 streaming (NT).
// ---------------------------------------------------------------------------
__global__ void score_kernel(const float* __restrict__ hs,
                             const float* __restrict__ v,
                             float* __restrict__ score) {
  const int gwave = (int)((blockIdx.x * blockDim.x + threadIdx.x) >> 5);
  const int lane  = threadIdx.x & 31;
  const v4f* row = (const v4f*)(hs + (size_t)gwave * H_DIM);
  const v4f* vv  = (const v4f*)v;
  float acc = 0.f;
#pragma unroll 4
  for (int t = lane; t < H_DIM / 4; t += 32) {
    v4f a = __builtin_nontemporal_load(row + t);
    v4f b = vv[t];
    acc = fmaf(a.x, b.x, fmaf(a.y, b.y, fmaf(a.z, b.z, fmaf(a.w, b.w, acc))));
  }
  for (int off = 16; off > 0; off >>= 1) acc += __shfl_xor(acc, off, 32);
  if (lane == 0) score[gwave] = acc;
}

// ---------------------------------------------------------------------------
// weights = softmax(score) over all S_DIM elements — single block of 1024.
// ---------------------------------------------------------------------------
__global__ void softmax_kernel(const float* __restrict__ score,
                               float* __restrict__ weights) {
  __shared__ float red[32];
  __shared__ float stat[2];
  const int tid  = threadIdx.x;        // 0..1023
  const int lane = tid & 31;
  const int wid  = tid >> 5;

  float vals[16];
  float m = -INFINITY;
#pragma unroll
  for (int i = 0; i < 16; ++i) {
    vals[i] = score[tid + i * 1024];
    m = fmaxf(m, vals[i]);
  }
  for (int off = 16; off > 0; off >>= 1) m = fmaxf(m, __shfl_xor(m, off, 32));
  if (lane == 0) red[wid] = m;
  __syncthreads();
  if (tid < 32) {
    float x = red[lane];
    for (int off = 16; off > 0; off >>= 1) x = fmaxf(x, __shfl_xor(x, off, 32));
    if (lane == 0) stat[0] = x;
  }
  __syncthreads();
  m = stat[0];

  float s = 0.f;
#pragma unroll
  for (int i = 0; i < 16; ++i) {
    vals[i] = expf(vals[i] - m);
    s += vals[i];
  }
  for (int off = 16; off > 0; off >>= 1) s += __shfl_xor(s, off, 32);
  if (lane == 0) red[wid] = s;
  __syncthreads();
  if (tid < 32) {
    float x = red[lane];
    for (int off = 16; off > 0; off >>= 1) x += __shfl_xor(x, off, 32);
    if (lane == 0) stat[1] = x;
  }
  __syncthreads();
  const float inv = 1.0f / stat[1];
#pragma unroll
  for (int i = 0; i < 16; ++i) weights[tid + i * 1024] = vals[i] * inv;
}

// ---------------------------------------------------------------------------
// out[i] = tanh(dot(W2[i, :], [context ; h_t]))  — one wave per output row.
// ---------------------------------------------------------------------------
__global__ void out_kernel(const float* __restrict__ W2,
                           const float* __restrict__ context,
                           const float* __restrict__ h_t,
                           float* __restrict__ out) {
  const int gwave = (int)((blockIdx.x * blockDim.x + threadIdx.x) >> 5);
  const int lane  = threadIdx.x & 31;
  const v4f* wrow = (const v4f*)(W2 + (size_t)gwave * (2 * H_DIM));
  const v4f* ctx4 = (const v4f*)context;
  const v4f* ht4  = (const v4f*)h_t;
  float acc = 0.f;
#pragma unroll 4
  for (int t = lane; t < (2 * H_DIM) / 4; t += 32) {
    v4f a = __builtin_nontemporal_load(wrow + t);
    v4f p = (t < H_DIM / 4) ? ctx4[t] : ht4[t - H_DIM / 4];
    acc = fmaf(a.x, p.x, fmaf(a.y, p.y, fmaf(a.z, p.z, fmaf(a.w, p.w, acc))));
  }
  for (int off = 16; off > 0; off >>= 1) acc += __shfl_xor(acc, off, 32);
  if (lane == 0) out[gwave] = tanhf(acc);
}

extern "C" void kernel_launch(void* const* d_in, const int* in_sizes, int n_in,
                              void* d_out, int out_size, void* d_ws, size_t ws_size,
                              hipStream_t stream) {
  (void)in_sizes; (void)n_in; (void)out_size; (void)ws_size;

  const float* hs = (const float*)d_in[0];   // (S, H)
  const float* W1 = (const float*)d_in[1];   // (H, H)
  const float* W2 = (const float*)d_in[2];   // (H, 2H)

  float* out_vec = (float*)d_out;            // attention_vector: H floats
  float* out_w   = (float*)d_out + H_DIM;    // attention_weights: S floats

  float* v       = (float*)d_ws;             // H floats
  float* score   = v + H_DIM;                // S floats
  float* context = score + S_DIM;            // H floats

  const float* h_t = hs + (size_t)(S_DIM - 1) * H_DIM;  // last row of hs

  // Zero the atomic accumulation targets (v, context; score is overwritten).
  hipMemsetAsync(d_ws, 0, (size_t)(H_DIM + S_DIM + H_DIM) * sizeof(float), stream);

  // 1) v = W1^T @ h_t   (WMMA column reduction over H rows of W1, 4 row-chunks)
  colreduce_wmma_kernel<<<dim3(H_DIM / 16, 4), 32, 0, stream>>>(
      W1, h_t, v, H_DIM / 4);

  // 2) score = hs @ v   (wave-per-row streaming dot)
  score_kernel<<<S_DIM / 8, 256, 0, stream>>>(hs, v, score);

  // 3) weights = softmax(score) -> directly into d_out
  softmax_kernel<<<1, 1024, 0, stream>>>(score, out_w);

  // 4) context = hs^T @ weights (WMMA column reduction over S rows, 16 chunks)
  colreduce_wmma_kernel<<<dim3(H_DIM / 16, 16), 32, 0, stream>>>(
      hs, out_w, context, S_DIM / 16);

  // 5) out_vec = tanh(W2 @ [context ; h_t])
  out_kernel<<<H_DIM / 8, 256, 0, stream>>>(W2, context, h_t, out_vec);
}